// RPNBaseModel_60636348285610
// MI455X (gfx1250) — compile-verified
//
#include <hip/hip_runtime.h>
#include <hip/hip_bf16.h>
#include <math.h>

#define GPAD 256          // padded GT count (LDS staging / tile width)
#define GCHUNK 32         // gts per LDS key-tile chunk
#define TPAD 257          // padded tile row stride (u64) -> conflict-free transpose read

typedef unsigned long long u64;

// -------------------------------------------------------------------------
// Kernel A: zero pos_from_gt flags and the num_positives output slot.
// -------------------------------------------------------------------------
__global__ __launch_bounds__(256) void rpn_init_kernel(int* __restrict__ posflag,
                                                       int* __restrict__ numpos, int A) {
    int a = blockIdx.x * 256 + threadIdx.x;
    if (a < A) posflag[a] = 0;
    if (a == 0) *numpos = 0;
}

// -------------------------------------------------------------------------
// Kernel B (fused): single IoU pass.
//  - GT boxes/classes async-DMA'd to LDS (CDNA5 global_load_async_to_lds)
//  - per-anchor row max / first-argmax / crowd-max  (registers)
//  - per-GT column argmax within the block via LDS key tile + transpose
//    reduction; one u64 partial per (block, gt) written to workspace.
//  key = (iou_bits << 32) | ~anchor_idx : iou>=0 so float bits are monotone,
//  ~idx gives jnp.argmax first-max tie-break; key==0 means "no candidate".
// -------------------------------------------------------------------------
__global__ __launch_bounds__(256) void rpn_fused_kernel(const float* __restrict__ anchors,
                                                        const unsigned char* __restrict__ valid,
                                                        const float* __restrict__ gt_boxes,
                                                        const int*   __restrict__ gt_cls,
                                                        float* __restrict__ rowmax,
                                                        int*   __restrict__ rowarg,
                                                        float* __restrict__ crowdm,
                                                        u64*   __restrict__ partials,
                                                        int A, int G) {
    __shared__ __align__(16) float4 s_gt[GPAD];
    __shared__ __align__(16) int    s_cls[GPAD];
    __shared__ float                s_area[GPAD];
    __shared__ u64                  s_keys[GCHUNK * TPAD];  // 32 x 257 u64 = 64.25 KB
    __shared__ u64                  s_pbest[GCHUNK * 8];    // 8 wave-parts per gt

    const int t = threadIdx.x;

    // ---- async DMA of GT table: one 16B box + one 4B class id per lane ----
    if (t < G) {
        unsigned lds_box = (unsigned)(size_t)(&s_gt[t]);
        unsigned lds_cls = (unsigned)(size_t)(&s_cls[t]);
        unsigned off16   = (unsigned)t * 16u;
        unsigned off4    = (unsigned)t * 4u;
        asm volatile("global_load_async_to_lds_b128 %0, %1, %2"
                     :: "v"(lds_box), "v"(off16), "s"(gt_boxes) : "memory");
        asm volatile("global_load_async_to_lds_b32 %0, %1, %2"
                     :: "v"(lds_cls), "v"(off4), "s"(gt_cls) : "memory");
    }
    asm volatile("s_wait_asynccnt 0x0" ::: "memory");
    __syncthreads();

    if (t >= G) {                        // neutral padding: crowd + unit box
        s_gt[t]  = make_float4(0.f, 0.f, 1.f, 1.f);
        s_cls[t] = -1;
    }
    {
        float4 g = s_gt[t];              // own write (pad) or post-barrier DMA data
        s_area[t] = (g.z - g.x) * (g.w - g.y);
    }
    __syncthreads();

    const int  a      = blockIdx.x * 256 + t;
    const bool inA    = a < A;
    float4 ab         = inA ? ((const float4*)anchors)[a] : make_float4(0.f, 0.f, 1.f, 1.f);
    const bool v      = inA && (valid[a] != 0);
    const float aArea = (ab.z - ab.x) * (ab.w - ab.y);

    float best  = -1e30f;
    int   bestg = 0;
    float cmax  = 0.0f;

    const int gl_r   = t & 31;           // gt lane for transpose reduction
    const int part_r = t >> 5;           // which 32-anchor slice this thread reduces

    for (int c = 0; c < GPAD / GCHUNK; ++c) {
        const int base = c * GCHUNK;

        #pragma unroll 8
        for (int gl = 0; gl < GCHUNK; ++gl) {
            const int gi = base + gl;
            float4 gb = s_gt[gi];                       // LDS broadcast read
            float y1 = fmaxf(ab.x, gb.x);
            float x1 = fmaxf(ab.y, gb.y);
            float y2 = fminf(ab.z, gb.z);
            float x2 = fminf(ab.w, gb.w);
            float inter = fmaxf(y2 - y1, 0.0f) * fmaxf(x2 - x1, 0.0f);
            float iou   = inter / (aArea + s_area[gi] - inter);
            bool  crowd = s_cls[gi] < 0;
            float ov    = crowd ? -1.0f : iou;
            if (ov > best) { best = ov; bestg = gi; }   // strict > => first argmax
            cmax = fmaxf(cmax, crowd ? iou : 0.0f);

            u64 key = 0ull;
            if (v && !crowd)
                key = ((u64)__float_as_uint(iou) << 32) |
                      (u64)(0xFFFFFFFFu - (unsigned)a);
            s_keys[gl * TPAD + t] = key;
        }
        __syncthreads();

        // transpose reduction: 8 threads per gt, each folds a 32-anchor slice
        {
            u64 m = 0ull;
            const u64* row = &s_keys[gl_r * TPAD + part_r * 32];
            #pragma unroll 8
            for (int i = 0; i < 32; ++i) { u64 k = row[i]; if (k > m) m = k; }
            s_pbest[gl_r * 8 + part_r] = m;
        }
        __syncthreads();

        if (t < GCHUNK) {
            u64 m = 0ull;
            #pragma unroll
            for (int i = 0; i < 8; ++i) { u64 k = s_pbest[t * 8 + i]; if (k > m) m = k; }
            partials[(size_t)blockIdx.x * GPAD + base + t] = m;   // coalesced
        }
        __syncthreads();   // protect s_keys/s_pbest before next chunk
    }

    if (inA) {
        rowmax[a] = best;
        rowarg[a] = bestg;
        crowdm[a] = cmax;
    }
}

// -------------------------------------------------------------------------
// Kernel C: fold per-block column partials -> pos_from_gt scatter.
// One block, thread g folds over all anchor-blocks (coalesced, prefetched).
// Crowd columns produced only zero keys, so best==0 => no scatter (matches
// the reference's `.max(~is_crowd)` no-op for crowd gts).
// -------------------------------------------------------------------------
__global__ __launch_bounds__(256) void rpn_colreduce_kernel(const u64* __restrict__ partials,
                                                            int* __restrict__ posflag,
                                                            int nblocks) {
    const int g = threadIdx.x;
    u64 best = 0ull;
    for (int b = 0; b < nblocks; ++b) {
        __builtin_prefetch(&partials[(size_t)(b + 8) * GPAD + g], 0, 0);
        u64 k = partials[(size_t)b * GPAD + g];
        if (k > best) best = k;
    }
    if (best != 0ull) {
        unsigned aw = 0xFFFFFFFFu - (unsigned)(best & 0xFFFFFFFFull);
        posflag[aw] = 1;                 // value-1 write races are benign
    }
}

// -------------------------------------------------------------------------
// Kernel D: finalize. match / deltas / positive count.
// -------------------------------------------------------------------------
__global__ __launch_bounds__(256) void rpn_fin_kernel(const float* __restrict__ anchors,
                                                      const unsigned char* __restrict__ valid,
                                                      const float* __restrict__ gt_boxes,
                                                      const float* __restrict__ stddev,
                                                      const float* __restrict__ rowmax,
                                                      const int*   __restrict__ rowarg,
                                                      const float* __restrict__ crowdm,
                                                      const int*   __restrict__ posflag,
                                                      int*   __restrict__ match,
                                                      float* __restrict__ bbox,
                                                      int*   __restrict__ numpos, int A) {
    __shared__ int cnt;
    if (threadIdx.x == 0) cnt = 0;
    __syncthreads();

    const int a = blockIdx.x * 256 + threadIdx.x;
    if (a < A) {
        float amax    = rowmax[a];
        int   ag      = rowarg[a];
        bool  nocrowd = crowdm[a] < 0.001f;   // any_crowd branch is redundant (max==0 w/o crowds)
        bool  v       = valid[a] != 0;
        bool  pos     = (posflag[a] != 0) || (amax >= 0.7f);
        bool  neg     = (amax < 0.3f) && nocrowd && !pos;
        match[a] = v ? (pos ? 1 : (neg ? -1 : 0)) : 0;

        bool pv = pos && v;
        float4 mg = ((const float4*)gt_boxes)[ag];
        float4 ab = ((const float4*)anchors)[a];
        // faithful to source, including the `- -` -> `+` size bug
        float gsy = mg.z + mg.x, gsx = mg.w + mg.y;
        float gcy = (mg.x + mg.z) * 0.5f, gcx = (mg.y + mg.w) * 0.5f;
        float asy = ab.z + ab.x, asx = ab.w + ab.y;
        float acy = (ab.x + ab.z) * 0.5f, acx = (ab.y + ab.w) * 0.5f;
        float d0 = ((gcy - acy) / asy) / stddev[0];
        float d1 = ((gcx - acx) / asx) / stddev[1];
        float d2 = logf(gsy / asy) / stddev[2];
        float d3 = logf(gsx / asx) / stddev[3];
        float4 o = pv ? make_float4(d0, d1, d2, d3) : make_float4(0.f, 0.f, 0.f, 0.f);
        ((float4*)bbox)[a] = o;
        if (pv) atomicAdd(&cnt, 1);
    }
    __syncthreads();
    if (threadIdx.x == 0 && cnt > 0) atomicAdd(numpos, cnt);
}

// -------------------------------------------------------------------------
extern "C" void kernel_launch(void* const* d_in, const int* in_sizes, int n_in,
                              void* d_out, int out_size, void* d_ws, size_t ws_size,
                              hipStream_t stream) {
    const float*         anchors = (const float*)d_in[0];
    const unsigned char* valid   = (const unsigned char*)d_in[1];  // jnp bool -> 1 byte
    const int*           gt_cls  = (const int*)d_in[2];
    const float*         gt_box  = (const float*)d_in[3];
    const float*         stddev  = (const float*)d_in[4];

    const int A = in_sizes[0] / 4;   // 261888
    int G = in_sizes[2];             // 256
    if (G > GPAD) G = GPAD;

    const int nblocks = (A + 255) / 256;   // 1023

    // workspace layout (16B aligned blocks)
    float* rowmax   = (float*)d_ws;                                   // A floats
    int*   rowarg   = (int*)d_ws + A;                                 // A ints
    float* crowdm   = (float*)d_ws + 2 * (size_t)A;                   // A floats
    int*   posflag  = (int*)d_ws + 3 * (size_t)A;                     // A ints
    size_t p_off    = ((4 * (size_t)A * sizeof(int)) + 15) & ~(size_t)15;
    u64*   partials = (u64*)((char*)d_ws + p_off);                    // nblocks*256 u64

    // output layout: [0,A) match(int32 bits), [A,5A) bbox(float), [5A] count(int32 bits)
    int*   match  = (int*)d_out;
    float* bbox   = (float*)d_out + A;
    int*   numpos = (int*)d_out + 5 * (size_t)A;

    rpn_init_kernel<<<nblocks, 256, 0, stream>>>(posflag, numpos, A);
    rpn_fused_kernel<<<nblocks, 256, 0, stream>>>(anchors, valid, gt_box, gt_cls,
                                                  rowmax, rowarg, crowdm, partials, A, G);
    rpn_colreduce_kernel<<<1, 256, 0, stream>>>(partials, posflag, nblocks);
    rpn_fin_kernel<<<nblocks, 256, 0, stream>>>(anchors, valid, gt_box, stddev,
                                                rowmax, rowarg, crowdm, posflag,
                                                match, bbox, numpos, A);
}